// CatAttention_46213848105806
// MI455X (gfx1250) — compile-verified
//
#include <hip/hip_runtime.h>
#include <hip/hip_bf16.h>
#include <math.h>

// Problem sizes (fixed by the reference)
#define B_  64
#define S_  2048
#define D_  128
#define DV_ 256
#define ROWS_PER_BLOCK 256
#define NCHUNK (S_ / ROWS_PER_BLOCK)   // 8 row-chunks per batch

// K3 LDS staging geometry
#define VROWS  16                       // V rows per stage
#define NSTAGE (ROWS_PER_BLOCK / VROWS) // 16 stages per chunk
#define VPAD   16                       // floats of row padding (bank-conflict free)
#define VLDROW (DV_ + VPAD)             // 272 floats per staged row

typedef __attribute__((ext_vector_type(2))) float v2f;
typedef __attribute__((ext_vector_type(4))) float v4f;
typedef __attribute__((ext_vector_type(8))) float v8f;
typedef __attribute__((ext_vector_type(4))) int   v4i;

// gfx1250 native v_tanh_f32 (transcendental unit); fall back to libm.
#if __has_builtin(__builtin_amdgcn_tanhf)
#define FAST_TANH(x) __builtin_amdgcn_tanhf(x)
#else
#define FAST_TANH(x) tanhf(x)
#endif

// gfx1250 async global->LDS copies (ASYNCcnt-tracked); guarded fallback.
#if __has_builtin(__builtin_amdgcn_global_load_async_to_lds_b128) && \
    __has_builtin(__builtin_amdgcn_s_wait_asynccnt)
#define HAVE_ASYNC 1
typedef __attribute__((address_space(1))) v4i* gptr_v4i;   // global b128 ptr
typedef __attribute__((address_space(3))) v4i* lptr_v4i;   // LDS b128 ptr
#else
#define HAVE_ASYNC 0
#endif

// ---------- wave32 reductions ----------
__device__ __forceinline__ float wave_sum(float v) {
#pragma unroll
  for (int o = 16; o > 0; o >>= 1) v += __shfl_xor(v, o, 32);
  return v;
}
__device__ __forceinline__ float wave_max(float v) {
#pragma unroll
  for (int o = 16; o > 0; o >>= 1) v = fmaxf(v, __shfl_xor(v, o, 32));
  return v;
}
__device__ __forceinline__ float block_max(float v, float* red) {
  const int wave = threadIdx.x >> 5, lane = threadIdx.x & 31;
  v = wave_max(v);
  __syncthreads();
  if (lane == 0) red[wave] = v;
  __syncthreads();
  float r = red[0];
#pragma unroll
  for (int i = 1; i < 8; ++i) r = fmaxf(r, red[i]);
  return r;
}
__device__ __forceinline__ float block_sum(float v, float* red) {
  const int wave = threadIdx.x >> 5, lane = threadIdx.x & 31;
  v = wave_sum(v);
  __syncthreads();
  if (lane == 0) red[wave] = v;
  __syncthreads();
  float r = red[0];
#pragma unroll
  for (int i = 1; i < 8; ++i) r += red[i];
  return r;
}

// ---------------------------------------------------------------------------
// K1: scores[b,s] = 0.5 * sum_d( tanh(q[b,s,d])*w_v[d] + tanh(k[b,s,d])*w_v[D+d] )
// (Path-2 of the reference is softmax over a length-1 axis => constant 1.0;
//  the +0.5 constant is dropped: softmax is shift-invariant.)
// One wave per row; each lane handles 4 contiguous floats of q and k (b128).
// ---------------------------------------------------------------------------
__global__ void __launch_bounds__(256)
scores_kernel(const float* __restrict__ q, const float* __restrict__ k,
              const float* __restrict__ wv, float* __restrict__ scores) {
  const int b     = blockIdx.x / NCHUNK;
  const int chunk = blockIdx.x % NCHUNK;
  const int wave  = threadIdx.x >> 5;
  const int lane  = threadIdx.x & 31;

  const v4f wq = *(const v4f*)(wv + lane * 4);
  const v4f wk = *(const v4f*)(wv + D_ + lane * 4);

  const float* qb = q + (size_t)b * S_ * D_;
  const float* kb = k + (size_t)b * S_ * D_;
  const int rows_per_wave = ROWS_PER_BLOCK / 8;        // 32 rows per wave

#pragma unroll 4
  for (int i = 0; i < rows_per_wave; ++i) {
    const int row = chunk * ROWS_PER_BLOCK + wave * rows_per_wave + i;
    const v4f qv = *(const v4f*)(qb + (size_t)row * D_ + lane * 4);
    const v4f kv = *(const v4f*)(kb + (size_t)row * D_ + lane * 4);
    float acc = 0.f;
    acc = fmaf(FAST_TANH(qv.x), wq.x, acc);
    acc = fmaf(FAST_TANH(qv.y), wq.y, acc);
    acc = fmaf(FAST_TANH(qv.z), wq.z, acc);
    acc = fmaf(FAST_TANH(qv.w), wq.w, acc);
    acc = fmaf(FAST_TANH(kv.x), wk.x, acc);
    acc = fmaf(FAST_TANH(kv.y), wk.y, acc);
    acc = fmaf(FAST_TANH(kv.z), wk.z, acc);
    acc = fmaf(FAST_TANH(kv.w), wk.w, acc);
    acc = wave_sum(acc);
    if (lane == 0) scores[(size_t)b * S_ + row] = 0.5f * acc;
  }
}

// ---------------------------------------------------------------------------
// K2: attn = softmax( where(mask, softmax(scores), -1e6) )  (both over S)
// The mask sits BETWEEN the softmaxes, so both are required. exp(-1e6 - m2)
// underflows to exactly 0 in f32. Also zero-initializes d_out each call.
// ---------------------------------------------------------------------------
__global__ void __launch_bounds__(256)
softmax2_kernel(const float* __restrict__ scores, const int* __restrict__ vlens,
                float* __restrict__ attn, float* __restrict__ out) {
  __shared__ float red[8];
  const int b = blockIdx.x, t = threadIdx.x;
  const int vlen = vlens[b];

  out[b * DV_ + t] = 0.f;   // 256 threads == DV_ columns

  float loc[8];
#pragma unroll
  for (int j = 0; j < 8; ++j) loc[j] = scores[(size_t)b * S_ + t + j * 256];

  float m = -__builtin_inff();
#pragma unroll
  for (int j = 0; j < 8; ++j) m = fmaxf(m, loc[j]);
  m = block_max(m, red);
  float sum = 0.f;
#pragma unroll
  for (int j = 0; j < 8; ++j) { loc[j] = __expf(loc[j] - m); sum += loc[j]; }
  sum = block_sum(sum, red);
  const float inv = 1.0f / sum;
#pragma unroll
  for (int j = 0; j < 8; ++j) loc[j] *= inv;

  float m2 = -1e6f;
#pragma unroll
  for (int j = 0; j < 8; ++j) {
    const int idx = t + j * 256;
    loc[j] = (idx < vlen) ? loc[j] : -1e6f;
    m2 = fmaxf(m2, loc[j]);
  }
  m2 = block_max(m2, red);
  float sum2 = 0.f;
#pragma unroll
  for (int j = 0; j < 8; ++j) {
    const int idx = t + j * 256;
    loc[j] = (idx < vlen) ? __expf(loc[j] - m2) : 0.f;
    sum2 += loc[j];
  }
  sum2 = block_sum(sum2, red);
  const float inv2 = 1.0f / sum2;
#pragma unroll
  for (int j = 0; j < 8; ++j)
    attn[(size_t)b * S_ + t + j * 256] = loc[j] * inv2;
}

// ---------------------------------------------------------------------------
// K3: out[b,:] += sum_s attn[b,s] * V[b,s,:]  via V_WMMA_F32_16X16X4_F32,
// fed from LDS tiles streamed in with async global->LDS b128 copies
// (ASYNCcnt double-buffering). Fragment layouts per ISA 7.12.2:
//   A 16x4: lanes 0-15 {K0,K1}, lanes 16-31 {K2,K3}  (attn broadcast down M)
//   B 4x16: same K striping, N = lane&15              (V tile rows)
//   C VGPR0 lanes 0-15 = (M=0, N) = the output tile.
// Row padding of 16 floats makes the two half-wave B reads hit disjoint
// bank groups ({c0..c0+15} vs {c0+32..c0+47} mod 64).
// attn[s]==0 exactly for s>=valid_len -> whole chunks past valid_len skipped.
// ---------------------------------------------------------------------------
__global__ void __launch_bounds__(256)
av_wmma_kernel(const float* __restrict__ attn, const float* __restrict__ values,
               const int* __restrict__ vlens, float* __restrict__ out) {
  const int b     = blockIdx.x / NCHUNK;
  const int chunk = blockIdx.x % NCHUNK;
  const int r0    = chunk * ROWS_PER_BLOCK;
  if (r0 >= vlens[b]) return;                 // uniform per block: EXEC stays full

  __shared__ float a_lds[ROWS_PER_BLOCK];
  __shared__ float v_lds[2][VROWS * VLDROW];  // 2 x 16 x 272 floats = 34 KB

  const int t = threadIdx.x;
  a_lds[t] = attn[(size_t)b * S_ + r0 + t];

  const float* vb = values + (size_t)b * S_ * DV_ + (size_t)r0 * DV_;

  // One stage = 16 rows x 256 cols = 1024 float4; 4 b128 copies per thread.
  // Lanes of a wave cover 512 contiguous bytes of global per copy.
  auto issue_stage = [&](int stage, int buf) {
    const float* src = vb + (size_t)stage * VROWS * DV_;
#pragma unroll
    for (int j = 0; j < 4; ++j) {
      const int idx  = t + j * 256;        // float4 index within stage
      const int row  = idx >> 6;           // DV_/4 = 64 float4 per row
      const int col4 = idx & 63;
      const float* gp = src + row * DV_ + col4 * 4;
      float* lp = &v_lds[buf][row * VLDROW + col4 * 4];
#if HAVE_ASYNC
      __builtin_amdgcn_global_load_async_to_lds_b128(
          (gptr_v4i)(gp), (lptr_v4i)(lp), /*offset=*/0, /*cpol=*/0);
#else
      *(v4f*)lp = *(const v4f*)gp;
#endif
    }
  };

  const int wave  = t >> 5;
  const int lane  = t & 31;
  const int half  = lane >> 4;              // 0: rows K0/K1, 1: rows K2/K3
  const int nlane = lane & 15;
  const int c0    = wave * 32;              // this wave's 32-column slice
  const float* vl0 = &v_lds[0][0];
  const float* vl1 = &v_lds[1][0];

  v8f acc0 = {}, acc1 = {};

  issue_stage(0, 0);
#pragma unroll 2
  for (int st = 0; st < NSTAGE; ++st) {
    const int buf = st & 1;
    if (st + 1 < NSTAGE) issue_stage(st + 1, buf ^ 1);
#if HAVE_ASYNC
    // Our 4 copies for stage st are complete once only the next stage's 4
    // remain in flight. Builtin requires a literal immediate.
    if (st + 1 < NSTAGE) {
      __builtin_amdgcn_s_wait_asynccnt(4);
    } else {
      __builtin_amdgcn_s_wait_asynccnt(0);
    }
#endif
    __syncthreads();                        // publish stage st (and a_lds on st==0)

    const float* vl = buf ? vl1 : vl0;
#pragma unroll
    for (int s = 0; s < VROWS; s += 4) {
      const int ai = st * VROWS + s + 2 * half;
      v2f A;  A.x = a_lds[ai];       A.y = a_lds[ai + 1];      // LDS broadcast
      const int rb = (s + 2 * half) * VLDROW + nlane;
      v2f B0; B0.x = vl[rb + c0];       B0.y = vl[rb + VLDROW + c0];
      v2f B1; B1.x = vl[rb + c0 + 16];  B1.y = vl[rb + VLDROW + c0 + 16];
      acc0 = __builtin_amdgcn_wmma_f32_16x16x4_f32(false, A, false, B0,
                                                   (short)0, acc0, false, false);
      acc1 = __builtin_amdgcn_wmma_f32_16x16x4_f32(false, A, false, B1,
                                                   (short)0, acc1, false, false);
    }
    __syncthreads();                        // all waves done reading buf before reuse
  }

  if (lane < 16) {                          // M=0 row of C = output tile
    atomicAdd(&out[b * DV_ + c0 + lane],      acc0[0]);
    atomicAdd(&out[b * DV_ + c0 + 16 + lane], acc1[0]);
  }
}

// ---------------------------------------------------------------------------
extern "C" void kernel_launch(void* const* d_in, const int* in_sizes, int n_in,
                              void* d_out, int out_size, void* d_ws, size_t ws_size,
                              hipStream_t stream) {
  const float* q     = (const float*)d_in[0];
  const float* k     = (const float*)d_in[1];
  const float* v     = (const float*)d_in[2];
  const int*   vlens = (const int*)  d_in[3];
  const float* wv    = (const float*)d_in[4];
  // d_in[5..7] (w2, w_v2_w, w_v2_b): dead path — softmax over a length-1 axis
  // is identically 1.0, and adding 0.5 is removed by softmax shift-invariance.
  float* out    = (float*)d_out;
  float* scores = (float*)d_ws;              // B_*S_ floats
  float* attn   = scores + (size_t)B_ * S_;  // B_*S_ floats

  scores_kernel  <<<dim3(B_ * NCHUNK), dim3(256), 0, stream>>>(q, k, wv, scores);
  softmax2_kernel<<<dim3(B_),          dim3(256), 0, stream>>>(scores, vlens, attn, out);
  av_wmma_kernel <<<dim3(B_ * NCHUNK), dim3(256), 0, stream>>>(attn, v, vlens, out);
}